// RealProteinStructureGNN_74259984548099
// MI455X (gfx1250) — compile-verified
//
#include <hip/hip_runtime.h>
#include <hip/hip_bf16.h>

// ---------------- problem constants (match reference) ----------------
constexpr int N_NODES = 100000;
constexpr int E_EDGES = 640000;
constexpr int F_IN    = 20;
constexpr int H_DIM   = 128;
constexpr int G_GR    = 256;
constexpr float EPS   = 1e-5f;

typedef float v2f __attribute__((ext_vector_type(2)));
typedef float v8f __attribute__((ext_vector_type(8)));
typedef unsigned int v4u __attribute__((ext_vector_type(4)));
typedef int v4i __attribute__((ext_vector_type(4)));
typedef int v8i __attribute__((ext_vector_type(8)));

// ---------------- utility ----------------
__global__ void zero_f32(float* __restrict__ p, int n) {
    int i = blockIdx.x * blockDim.x + threadIdx.x;
    if (i < n) p[i] = 0.0f;
}

// ---------------- degree / normalization ----------------
__global__ void deg_kernel(const int* __restrict__ dst, float* __restrict__ deg, int e) {
    int i = blockIdx.x * blockDim.x + threadIdx.x;
    if (i < e) atomicAdd(&deg[dst[i]], 1.0f);
}

__global__ void dis_kernel(const float* __restrict__ deg, float* __restrict__ dis, int n) {
    int i = blockIdx.x * blockDim.x + threadIdx.x;
    if (i < n) dis[i] = rsqrtf(deg[i] + 1.0f);   // +1 self loop; deg >= 1 always
}

// ---------------- LDS-staged WMMA GEMM for the layer matmuls ----------------
// C[M,128] = A[M,K] @ Bw[K,128].  Block = 256 threads = 8 waves; block covers 128 rows.
// Wave w computes rows [tm, tm+16) x all 128 cols = 8 WMMA tiles per k-step,
// reusing one A-fragment across all 8 tiles.  Bw (<= 64KB) is staged into LDS
// once per block by the Tensor Data Mover (wave 0 issues tensor_load_to_lds,
// waits on TENSORcnt, then the workgroup barrier releases everyone).
__global__ __launch_bounds__(256)
void wmma_gemm_lds(const float* __restrict__ A, const float* __restrict__ Bw,
                   float* __restrict__ C, int M, int K)
{
    __shared__ float smem[H_DIM * H_DIM];   // 64 KB max (K<=128) out of 320 KB/WGP

    const int wave = threadIdx.x >> 5;
    const int lane = threadIdx.x & 31;
    const int half = lane >> 4;    // 0 | 1
    const int l16  = lane & 15;

    // ---- stage Bw (K x 128 f32, row major) into LDS ----
    if (wave == 0) {
#if __has_builtin(__builtin_amdgcn_tensor_load_to_lds) && __has_builtin(__builtin_amdgcn_s_wait_tensorcnt)
        // Build Tensor DMA Descriptor (D#), CDNA5 ISA.
        unsigned long long ga = (unsigned long long)(uintptr_t)Bw;
        unsigned int lds_base = (unsigned int)(uintptr_t)smem;  // low 32b = LDS byte offset
        v4u g0;
        g0.x = 1u;                                               // count=1, valid, user mode
        g0.y = lds_base;                                         // lds_addr
        g0.z = (unsigned int)(ga & 0xFFFFFFFFull);               // global_addr[31:0]
        g0.w = (unsigned int)((ga >> 32) & 0x01FFFFFFull)        // global_addr[56:32]
             | (2u << 30);                                       // type = 2 ("image")
        v8i g1;
        g1[0] = (2 << 16);                                       // data_size = 4 bytes
        g1[1] = (H_DIM & 0xFFFF) << 16;                          // tensor_dim0[15:0] = 128
        g1[2] = (K & 0xFFFF) << 16;                              // dim0 hi=0 | tensor_dim1 lo = K
        g1[3] = (H_DIM & 0xFFFF) << 16;                          // dim1 hi=0 | tile_dim0 = 128
        g1[4] = (K & 0xFFFF);                                    // tile_dim1 = K, tile_dim2 = 0
        g1[5] = H_DIM;                                           // tensor_dim0_stride = 128
        g1[6] = 0;                                               // stride0 hi | stride1 lo
        g1[7] = 0;                                               // stride1 hi
        v4i gz4 = {0, 0, 0, 0};
        v8i gz8 = {0, 0, 0, 0, 0, 0, 0, 0};
        // amdgpu-toolchain (clang-23) 6-arg form:
        // (uint32x4 g0, int32x8 g1, int32x4, int32x4, int32x8, i32 cpol)
        __builtin_amdgcn_tensor_load_to_lds(g0, g1, gz4, gz4, gz8, 0);
        __builtin_amdgcn_s_wait_tensorcnt(0);                    // s_wait_tensorcnt 0
#else
        for (int i = lane; i < K * H_DIM; i += 32) smem[i] = Bw[i];
#endif
    }
    __syncthreads();

    const int tm = blockIdx.x * 128 + wave * 16;
    if (tm >= M) return;                     // uniform per wave

    v8f acc[8];
#pragma unroll
    for (int t = 0; t < 8; ++t)
        acc[t] = v8f{0.f, 0.f, 0.f, 0.f, 0.f, 0.f, 0.f, 0.f};

    // A-frag layout (V_WMMA_F32_16X16X4_F32): lane l, vgpr j -> A[M=l%16][K = j + 2*(l/16)]
    const float* arow = A + (size_t)(tm + l16) * K + 2 * half;
    for (int k = 0; k < K; k += 4) {
        v2f a;
        a.x = arow[k + 0];
        a.y = arow[k + 1];
        const float* brow0 = smem + (size_t)(k + 2 * half) * H_DIM + l16;
        const float* brow1 = brow0 + H_DIM;
#pragma unroll
        for (int t = 0; t < 8; ++t) {
            v2f b;                       // B[K = j + 2*half][N = t*16 + l16] from LDS
            b.x = brow0[t * 16];
            b.y = brow1[t * 16];
            acc[t] = __builtin_amdgcn_wmma_f32_16x16x4_f32(
                false, a, false, b, (short)0, acc[t], false, false);
        }
    }

    // C/D layout: lane l, vgpr v -> C[M = v + 8*(l/16)][N = l16]
    float* crow = C + (size_t)(tm + 8 * half) * H_DIM + l16;
#pragma unroll
    for (int t = 0; t < 8; ++t)
#pragma unroll
        for (int v = 0; v < 8; ++v)
            crow[(size_t)v * H_DIM + t * 16] = acc[t][v];
}

// ---------------- small generic WMMA GEMM (encoder head) ----------------
__global__ __launch_bounds__(32)
void wmma_gemm_f32(const float* __restrict__ A, const float* __restrict__ B,
                   const float* __restrict__ bias, float* __restrict__ C,
                   int M, int Nc, int K, int relu)
{
    const int lane = threadIdx.x & 31;
    const int half = lane >> 4;
    const int l16  = lane & 15;
    const int tn   = blockIdx.x * 16;
    const int tm   = blockIdx.y * 16;

    v8f acc = {0.f, 0.f, 0.f, 0.f, 0.f, 0.f, 0.f, 0.f};
    const float* arow = A + (size_t)(tm + l16) * K + 2 * half;
    const float* bcol = B + (size_t)(2 * half) * Nc + (tn + l16);

    for (int k = 0; k < K; k += 4) {
        v2f a, b;
        a.x = arow[k + 0];
        a.y = arow[k + 1];
        b.x = bcol[(size_t)(k + 0) * Nc];
        b.y = bcol[(size_t)(k + 1) * Nc];
        acc = __builtin_amdgcn_wmma_f32_16x16x4_f32(
            false, a, false, b, (short)0, acc, false, false);
    }

    const float bv = bias ? bias[tn + l16] : 0.0f;
    float* crow = C + (size_t)(tm + 8 * half) * Nc + (tn + l16);
#pragma unroll
    for (int v = 0; v < 8; ++v) {
        float x = acc[v] + bv;
        if (relu) x = fmaxf(x, 0.0f);
        crow[(size_t)v * Nc] = x;
    }
}

// ---------------- GCN scatter: out = selfloop + bias, then edge atomics ----------------
__global__ void scat_init(const float* __restrict__ hw, const float* __restrict__ dis,
                          const float* __restrict__ bias, float* __restrict__ out, int nh)
{
    int i = blockIdx.x * blockDim.x + threadIdx.x;
    if (i >= nh) return;
    int r = i >> 7;            // / H_DIM
    int c = i & (H_DIM - 1);
    float d = dis[r];
    out[i] = hw[i] * d * d + bias[c];
}

// 32 lanes per edge, float4 per lane => one coalesced 512B row read per edge
__global__ void scat_edges(const float* __restrict__ hw, const float* __restrict__ dis,
                           const int* __restrict__ src, const int* __restrict__ dst,
                           float* __restrict__ out, int e)
{
    long long tid = (long long)blockIdx.x * blockDim.x + threadIdx.x;
    int ei = (int)(tid >> 5);
    if (ei >= e) return;
    int q  = (int)(tid & 31);
    // stream the index arrays ahead of the atomics (global_prefetch_b8)
    if (ei + 8192 < e && q == 0) {
        __builtin_prefetch(src + ei + 8192, 0, 0);
        __builtin_prefetch(dst + ei + 8192, 0, 0);
    }
    int s  = src[ei];
    int d  = dst[ei];
    float nrm = dis[s] * dis[d];
    const float4* hr = (const float4*)(hw + (size_t)s * H_DIM) + q;
    float4 v = *hr;
    float* o = out + (size_t)d * H_DIM + q * 4;
    atomicAdd(o + 0, v.x * nrm);
    atomicAdd(o + 1, v.y * nrm);
    atomicAdd(o + 2, v.z * nrm);
    atomicAdd(o + 3, v.w * nrm);
}

// ---------------- batchnorm ----------------
__global__ __launch_bounds__(H_DIM)
void bn_stats(const float* __restrict__ h, float* __restrict__ stats, int n)
{
    int c = threadIdx.x;                 // channel
    float s = 0.0f, s2 = 0.0f;
    for (int r = blockIdx.x; r < n; r += gridDim.x) {
        float v = h[(size_t)r * H_DIM + c];
        s += v; s2 += v * v;
    }
    atomicAdd(&stats[c], s);
    atomicAdd(&stats[H_DIM + c], s2);
}

__global__ void bn_apply(float* __restrict__ h, const float* __restrict__ stats,
                         const float* __restrict__ g, const float* __restrict__ beta, int nh)
{
    int i = blockIdx.x * blockDim.x + threadIdx.x;
    if (i >= nh) return;
    int c = i & (H_DIM - 1);
    const float inv_n = 1.0f / (float)N_NODES;
    float mean = stats[c] * inv_n;
    float var  = stats[H_DIM + c] * inv_n - mean * mean;
    float v = (h[i] - mean) * rsqrtf(var + EPS) * g[c] + beta[c];
    h[i] = fmaxf(v, 0.0f);
}

// ---------------- pooling ----------------
__global__ void cnt_kernel(const int* __restrict__ batch, float* __restrict__ cnts, int n) {
    int i = blockIdx.x * blockDim.x + threadIdx.x;
    if (i < n) atomicAdd(&cnts[batch[i]], 1.0f);
}

__global__ void pool_kernel(const float* __restrict__ h, const int* __restrict__ batch,
                            float* __restrict__ pool, int nh)
{
    int i = blockIdx.x * blockDim.x + threadIdx.x;
    if (i >= nh) return;
    int r = i >> 7;
    int c = i & (H_DIM - 1);
    atomicAdd(&pool[(size_t)batch[r] * H_DIM + c], h[i]);
}

// pbuf[g, 0:128] = pbuf[g, 128:256] = mean (reference concatenates mean with itself)
__global__ void mean_kernel(const float* __restrict__ pool, const float* __restrict__ cnts,
                            float* __restrict__ pbuf, int gh)
{
    int i = blockIdx.x * blockDim.x + threadIdx.x;
    if (i >= gh) return;
    int g = i >> 7;
    int c = i & (H_DIM - 1);
    float m = pool[i] / fmaxf(cnts[g], 1.0f);
    pbuf[(size_t)g * (2 * H_DIM) + c] = m;
    pbuf[(size_t)g * (2 * H_DIM) + H_DIM + c] = m;
}

// ---------------- launcher ----------------
extern "C" void kernel_launch(void* const* d_in, const int* in_sizes, int n_in,
                              void* d_out, int out_size, void* d_ws, size_t ws_size,
                              hipStream_t stream)
{
    (void)in_sizes; (void)n_in; (void)out_size; (void)ws_size;

    const float* x     = (const float*)d_in[0];
    const int*   ei    = (const int*)d_in[1];
    const int*   src   = ei;
    const int*   dst   = ei + E_EDGES;
    const int*   batch = (const int*)d_in[2];
    const float* w[3]  = {(const float*)d_in[3], (const float*)d_in[5], (const float*)d_in[7]};
    const float* b[3]  = {(const float*)d_in[4], (const float*)d_in[6], (const float*)d_in[8]};
    const float* gam[3]= {(const float*)d_in[9],  (const float*)d_in[11], (const float*)d_in[13]};
    const float* bet[3]= {(const float*)d_in[10], (const float*)d_in[12], (const float*)d_in[14]};
    const float* ew1   = (const float*)d_in[15];
    const float* eb1   = (const float*)d_in[16];
    const float* ew2   = (const float*)d_in[17];
    const float* eb2   = (const float*)d_in[18];

    // workspace layout (floats)
    float* ws    = (float*)d_ws;
    float* deg   = ws;                                 // N
    float* dis   = deg  + N_NODES;                     // N
    float* hA    = dis  + N_NODES;                     // N*H (gemm out)
    float* hB    = hA   + (size_t)N_NODES * H_DIM;     // N*H (scatter/bn out)
    float* stats = hB   + (size_t)N_NODES * H_DIM;     // 2*H
    float* pool  = stats + 2 * H_DIM;                  // G*H
    float* cnts  = pool + (size_t)G_GR * H_DIM;        // G
    float* pbuf  = cnts + G_GR;                        // G*2H
    float* ebuf  = pbuf + (size_t)G_GR * 2 * H_DIM;    // G*H

    const int NH = N_NODES * H_DIM;
    const int TB = 256;
    auto cdiv = [](long long a, long long b) { return (int)((a + b - 1) / b); };

    // degrees / dis
    zero_f32<<<cdiv(N_NODES, TB), TB, 0, stream>>>(deg, N_NODES);
    deg_kernel<<<cdiv(E_EDGES, TB), TB, 0, stream>>>(dst, deg, E_EDGES);
    dis_kernel<<<cdiv(N_NODES, TB), TB, 0, stream>>>(deg, dis, N_NODES);

    const float* hin = x;
    int K = F_IN;
    for (int l = 0; l < 3; ++l) {
        // hA = hin @ w[l]   (TDM-staged weight panel in LDS, 8 waves x 8 tiles)
        wmma_gemm_lds<<<cdiv(N_NODES, 128), 256, 0, stream>>>(hin, w[l], hA, N_NODES, K);
        // hB = selfloop + bias, then += edge messages
        scat_init<<<cdiv(NH, TB), TB, 0, stream>>>(hA, dis, b[l], hB, NH);
        scat_edges<<<cdiv((long long)E_EDGES * 32, TB), TB, 0, stream>>>(
            hA, dis, src, dst, hB, E_EDGES);
        // batchnorm + relu (in place on hB)
        zero_f32<<<1, 2 * H_DIM, 0, stream>>>(stats, 2 * H_DIM);
        bn_stats<<<512, H_DIM, 0, stream>>>(hB, stats, N_NODES);
        bn_apply<<<cdiv(NH, TB), TB, 0, stream>>>(hB, stats, gam[l], bet[l], NH);
        hin = hB;
        K = H_DIM;
    }

    // pooling
    zero_f32<<<cdiv(G_GR * H_DIM + G_GR, TB), TB, 0, stream>>>(pool, G_GR * H_DIM + G_GR);
    cnt_kernel<<<cdiv(N_NODES, TB), TB, 0, stream>>>(batch, cnts, N_NODES);
    pool_kernel<<<cdiv(NH, TB), TB, 0, stream>>>(hB, batch, pool, NH);
    mean_kernel<<<cdiv(G_GR * H_DIM, TB), TB, 0, stream>>>(pool, cnts, pbuf, G_GR * H_DIM);

    // encoder: ebuf = relu(pbuf[256,256] @ ew1[256,128] + eb1)
    wmma_gemm_f32<<<dim3(H_DIM / 16, G_GR / 16), 32, 0, stream>>>(
        pbuf, ew1, eb1, ebuf, G_GR, H_DIM, 2 * H_DIM, 1);
    // out = ebuf[256,128] @ ew2[128,64] + eb2
    wmma_gemm_f32<<<dim3((H_DIM / 2) / 16, G_GR / 16), 32, 0, stream>>>(
        ebuf, ew2, eb2, (float*)d_out, G_GR, H_DIM / 2, H_DIM, 0);
}